// HybridModel_88965952569393
// MI455X (gfx1250) — compile-verified
//
#include <hip/hip_runtime.h>
#include <hip/hip_bf16.h>

// ---------------------------------------------------------------------------
// Quanvolutional hybrid model, fully fused on-device.
//   d_in[0] = x        (512*28*28 f32)
//   d_in[1] = q_params (16 f32)
//   d_in[2] = W        (2*1521 f32)
//   d_in[3] = b        (2 f32)
//   d_out   = log_softmax logits (512*2 f32)
// ---------------------------------------------------------------------------

#define N_IMG    512
#define N_PATCH  169
#define N_WIRES  9
#define DIM      512          // 2^9
#define K_RAW    1521         // 169*9
#define K_PAD    1524         // multiple of 4 for wmma f32 16x16x4
#define WAVES_PER_BLOCK 8

__device__ __forceinline__ unsigned insertBit(unsigned v, int b, unsigned bit) {
    return ((v >> b) << (b + 1)) | (bit << b) | (v & ((1u << b) - 1u));
}

// TYPE: 0=rx, 1=ry, 2=rz.  BP = bit position of the wire (bp = 8 - wire).
template <int BP, int TYPE>
__device__ __forceinline__ void applyRot(float2* st, int lane, float c, float s) {
#pragma unroll
    for (int r = 0; r < 8; ++r) {
        unsigned pi = (unsigned)lane + 32u * (unsigned)r;   // 0..255 pair id
        unsigned i0 = insertBit(pi, BP, 0u);
        unsigned i1 = i0 | (1u << BP);
        float2 a0 = st[i0];
        float2 a1 = st[i1];
        float2 n0, n1;
        if (TYPE == 0) {            // rx: [[c,-is],[-is,c]]
            n0.x =  c * a0.x + s * a1.y;   n0.y =  c * a0.y - s * a1.x;
            n1.x =  s * a0.y + c * a1.x;   n1.y = -s * a0.x + c * a1.y;
        } else if (TYPE == 1) {     // ry: [[c,-s],[s,c]]
            n0.x =  c * a0.x - s * a1.x;   n0.y =  c * a0.y - s * a1.y;
            n1.x =  s * a0.x + c * a1.x;   n1.y =  s * a0.y + c * a1.y;
        } else {                    // rz: diag(e^{-i t/2}, e^{+i t/2})
            n0.x =  c * a0.x + s * a0.y;   n0.y =  c * a0.y - s * a0.x;
            n1.x =  c * a1.x - s * a1.y;   n1.y =  c * a1.y + s * a1.x;
        }
        st[i0] = n0;
        st[i1] = n1;
    }
    __builtin_amdgcn_wave_barrier();   // in-wave LDS ops are in-order; fence compiler only
}

// CNOT: control bit position BC, target bit position BT (final-index positions)
template <int BC, int BT>
__device__ __forceinline__ void applyCnot(float2* st, int lane) {
    constexpr int BLO = (BC < BT) ? BC : BT;
    constexpr int BHI = (BC < BT) ? BT : BC;
    constexpr unsigned VLO = (BLO == BC) ? 1u : 0u;  // control fixed to 1, target to 0
    constexpr unsigned VHI = (BHI == BC) ? 1u : 0u;
#pragma unroll
    for (int r = 0; r < 4; ++r) {
        unsigned pi = (unsigned)lane + 32u * (unsigned)r;   // 0..127
        unsigned i  = insertBit(pi, BLO, VLO);
        i           = insertBit(i,  BHI, VHI);
        unsigned i0 = i;                 // control=1, target=0
        unsigned i1 = i | (1u << BT);    // control=1, target=1
        float2 t = st[i0];
        st[i0] = st[i1];
        st[i1] = t;
    }
    __builtin_amdgcn_wave_barrier();
}

// One wave per patch; state vector (512 complex f32) lives in LDS.
__global__ void __launch_bounds__(256) quanv_kernel(const float* __restrict__ x,
                                                    const float* __restrict__ qp,
                                                    float* __restrict__ feats) {
    __shared__ float2 sState[WAVES_PER_BLOCK * DIM];   // 32 KB
    const int lane = threadIdx.x & 31;
    const int wid  = threadIdx.x >> 5;
    const int gw   = blockIdx.x * WAVES_PER_BLOCK + wid;   // exact: grid covers 86528
    const int b    = gw / N_PATCH;
    const int p    = gw - b * N_PATCH;
    const int py   = p / 13;
    const int px   = p - py * 13;
    float2* st = sState + wid * DIM;

    // ---- amplitude encoding: per-wire (cos(d/2), sin(d/2)) ----
    float cv = 1.0f, sv = 0.0f;
    if (lane < N_WIRES) {
        const int i = lane / 3;
        const int j = lane - 3 * i;
        const float d = x[b * 784 + (2 * py + i) * 28 + (2 * px + j)];
        const float h = 0.5f * d;
        cv = __cosf(h);
        sv = __sinf(h);
    }
    float cd[N_WIRES], sd[N_WIRES];
#pragma unroll
    for (int w = 0; w < N_WIRES; ++w) {
        cd[w] = __shfl(cv, w, 32);
        sd[w] = __shfl(sv, w, 32);
    }

    // ---- product state init (wire 0 = MSB of flat index) ----
#pragma unroll
    for (int j = 0; j < 16; ++j) {
        const unsigned idx = (unsigned)lane * 16u + (unsigned)j;
        float amp = 1.0f;
#pragma unroll
        for (int w = 0; w < N_WIRES; ++w)
            amp *= ((idx >> (8 - w)) & 1u) ? sd[w] : cd[w];
        st[idx] = make_float2(amp, 0.0f);
    }
    __builtin_amdgcn_wave_barrier();

    // ---- 16-gate circuit (wire w -> bit position 8-w) ----
    float th, c, s;
    th = qp[0];  c = __cosf(0.5f * th); s = __sinf(0.5f * th); applyRot<8, 0>(st, lane, c, s); // rx w0
    th = qp[1];  c = __cosf(0.5f * th); s = __sinf(0.5f * th); applyRot<7, 1>(st, lane, c, s); // ry w1
    th = qp[2];  c = __cosf(0.5f * th); s = __sinf(0.5f * th); applyRot<6, 2>(st, lane, c, s); // rz w2
    applyCnot<5, 4>(st, lane);                                                                 // cnot(3,4)
    th = qp[4];  c = __cosf(0.5f * th); s = __sinf(0.5f * th); applyRot<4, 0>(st, lane, c, s); // rx w4
    th = qp[5];  c = __cosf(0.5f * th); s = __sinf(0.5f * th); applyRot<3, 1>(st, lane, c, s); // ry w5
    th = qp[6];  c = __cosf(0.5f * th); s = __sinf(0.5f * th); applyRot<2, 2>(st, lane, c, s); // rz w6
    applyCnot<1, 0>(st, lane);                                                                 // cnot(7,8)
    th = qp[8];  c = __cosf(0.5f * th); s = __sinf(0.5f * th); applyRot<0, 0>(st, lane, c, s); // rx w8
    th = qp[9];  c = __cosf(0.5f * th); s = __sinf(0.5f * th); applyRot<8, 1>(st, lane, c, s); // ry w0
    th = qp[10]; c = __cosf(0.5f * th); s = __sinf(0.5f * th); applyRot<7, 2>(st, lane, c, s); // rz w1
    applyCnot<6, 5>(st, lane);                                                                 // cnot(2,3)
    th = qp[12]; c = __cosf(0.5f * th); s = __sinf(0.5f * th); applyRot<5, 0>(st, lane, c, s); // rx w3
    th = qp[13]; c = __cosf(0.5f * th); s = __sinf(0.5f * th); applyRot<4, 1>(st, lane, c, s); // ry w4
    th = qp[14]; c = __cosf(0.5f * th); s = __sinf(0.5f * th); applyRot<3, 2>(st, lane, c, s); // rz w5
    applyCnot<2, 1>(st, lane);                                                                 // cnot(6,7)

    // ---- Z expectations: feat[w] = sum_i |amp_i|^2 * (1 - 2*bit_w(i)) ----
    float z[N_WIRES];
#pragma unroll
    for (int w = 0; w < N_WIRES; ++w) z[w] = 0.0f;
#pragma unroll
    for (int j = 0; j < 16; ++j) {
        const unsigned idx = (unsigned)lane * 16u + (unsigned)j;
        const float2 a = st[idx];
        const float pr = a.x * a.x + a.y * a.y;
#pragma unroll
        for (int w = 0; w < N_WIRES; ++w)
            z[w] += ((idx >> (8 - w)) & 1u) ? -pr : pr;
    }
#pragma unroll
    for (int w = 0; w < N_WIRES; ++w) {
#pragma unroll
        for (int off = 16; off > 0; off >>= 1)
            z[w] += __shfl_xor(z[w], off, 32);
    }
    if (lane == 0) {
        float* dst = feats + b * K_PAD + p * N_WIRES;
#pragma unroll
        for (int w = 0; w < N_WIRES; ++w) dst[w] = z[w];
    }
}

// Build zero-padded W^T laid out as BT[n][k], n in 0..15, k in 0..K_PAD-1.
__global__ void build_b_kernel(const float* __restrict__ W, float* __restrict__ BT) {
    const int idx = blockIdx.x * 256 + threadIdx.x;
    if (idx >= 16 * K_PAD) return;
    const int n = idx / K_PAD;
    const int k = idx - n * K_PAD;
    BT[idx] = (n < 2 && k < K_RAW) ? W[n * K_RAW + k] : 0.0f;
}

// Zero the 3 K-padding columns of feats.
__global__ void zero_pad_kernel(float* __restrict__ feats) {
    const int idx = blockIdx.x * 256 + threadIdx.x;
    if (idx >= N_IMG * (K_PAD - K_RAW)) return;
    const int r = idx / (K_PAD - K_RAW);
    const int c = idx - r * (K_PAD - K_RAW);
    feats[r * K_PAD + K_RAW + c] = 0.0f;
}

// ---------------------------------------------------------------------------
// Head: logits = feats @ W^T + b, then log_softmax, via V_WMMA_F32_16X16X4_F32.
// One wave per 16 rows. A frag: lanes 0-15 -> K=0,1 ; lanes 16-31 -> K=2,3.
// B frag mirrors with N = lane%16. D: VGPR r holds row r + 8*(lane>=16), col lane%16.
// ---------------------------------------------------------------------------
typedef __attribute__((ext_vector_type(2))) float v2f;
typedef __attribute__((ext_vector_type(8))) float v8f;

__global__ void __launch_bounds__(32) head_kernel(const float* __restrict__ feats,
                                                  const float* __restrict__ BT,
                                                  const float* __restrict__ bias,
                                                  float* __restrict__ out) {
    const int lane = threadIdx.x;          // 32 threads, all active (EXEC all ones)
    const int row0 = blockIdx.x * 16;
    const int m    = lane & 15;
    const int kb   = (lane >> 4) << 1;     // 0 or 2
    const float* arow = feats + (row0 + m) * K_PAD + kb;
    const float* brow = BT + m * K_PAD + kb;

    v8f acc = {};
#pragma unroll 4
    for (int kk = 0; kk < K_PAD; kk += 4) {
        v2f a, bf;
        a.x  = arow[kk];
        a.y  = arow[kk + 1];
        bf.x = brow[kk];
        bf.y = brow[kk + 1];
        acc = __builtin_amdgcn_wmma_f32_16x16x4_f32(false, a, false, bf,
                                                    (short)0, acc, false, false);
    }

    // log-softmax over the 2 valid columns; lanes {0,1,16,17} own the outputs.
    const float b01 = bias[lane & 1];
#pragma unroll
    for (int r = 0; r < 8; ++r) {
        const float mine  = acc[r] + b01;
        const float other = __shfl_xor(mine, 1, 32);
        const float mx  = fmaxf(mine, other);
        const float lse = mx + __logf(__expf(mine - mx) + __expf(other - mx));
        if (m < 2) {
            const int row = row0 + r + ((lane >> 4) << 3);
            out[row * 2 + (lane & 1)] = mine - lse;
        }
    }
}

extern "C" void kernel_launch(void* const* d_in, const int* in_sizes, int n_in,
                              void* d_out, int out_size, void* d_ws, size_t ws_size,
                              hipStream_t stream) {
    const float* x  = (const float*)d_in[0];
    const float* qp = (const float*)d_in[1];
    const float* W  = (const float*)d_in[2];
    const float* b  = (const float*)d_in[3];
    float* feats = (float*)d_ws;                    // 512*1524 f32
    float* BT    = feats + (size_t)N_IMG * K_PAD;   // 16*1524 f32
    float* out   = (float*)d_out;

    build_b_kernel<<<(16 * K_PAD + 255) / 256, 256, 0, stream>>>(W, BT);
    zero_pad_kernel<<<(N_IMG * (K_PAD - K_RAW) + 255) / 256, 256, 0, stream>>>(feats);
    quanv_kernel<<<(N_IMG * N_PATCH) / WAVES_PER_BLOCK, 256, 0, stream>>>(x, qp, feats);
    head_kernel<<<N_IMG / 16, 32, 0, stream>>>(feats, BT, b, out);
}